// CompGCNConv_27178553049425
// MI455X (gfx1250) — compile-verified
//
#include <hip/hip_runtime.h>

typedef float v2f __attribute__((ext_vector_type(2)));
typedef float v8f __attribute__((ext_vector_type(8)));

#define D128 128
#define NEG_SLOPE 0.01f
#define BN_EPS 1e-5f

// ---------------- zero workspace ----------------
__global__ void k_zero(float* __restrict__ p, int n) {
  int i = blockIdx.x * blockDim.x + threadIdx.x;
  if (i < n) p[i] = 0.f;
}

// ---------------- weight prep: fold 0.5 and loop_rel into weights ------
// wA[k][n] = 0.5 * w_out[k][n]
// wB[k][n] = 0.5 * loop_rel[k] * w_loop[k][n]   ((x*l) @ W == x @ diag(l)W)
__global__ void k_wprep(const float* __restrict__ w_out,
                        const float* __restrict__ w_loop,
                        const float* __restrict__ loop_rel,
                        float* __restrict__ wA, float* __restrict__ wB) {
  int i = blockIdx.x * blockDim.x + threadIdx.x;  // 0 .. 16383
  if (i >= D128 * D128) return;
  int k = i >> 7;
  wA[i] = 0.5f * w_out[i];
  wB[i] = 0.5f * loop_rel[k] * w_loop[i];
}

// ---------------- per-relation precompute -------
// For each relation r (incl. loop rel at index n_rel):
//   relW[k] = rel_all[r] @ W_w[:,k] + W_b[k]
//   s1[r] = relW . a_w[0:128];  s2[r] = relW . a_w[128:256]
//   rel_out[r] = rel_all[r] @ w_rel   (only r < n_rel written)
__global__ __launch_bounds__(128) void k_rel_prep(
    const float* __restrict__ rel_embed, const float* __restrict__ loop_rel,
    const float* __restrict__ W_w, const float* __restrict__ W_b,
    const float* __restrict__ a_w, const float* __restrict__ w_rel,
    float* __restrict__ s1, float* __restrict__ s2,
    float* __restrict__ rel_out, int n_rel) {
  int r = blockIdx.x;          // 0 .. n_rel (n_rel+1 blocks)
  int k = threadIdx.x;         // 0 .. 127
  __shared__ float rv[128];
  __shared__ float red[128];
  const float* src = (r < n_rel) ? (rel_embed + (size_t)r * D128) : loop_rel;
  rv[k] = src[k];
  __syncthreads();
  float accW = 0.f, accR = 0.f;
#pragma unroll 8
  for (int j = 0; j < D128; ++j) {
    float e = rv[j];
    accW += e * W_w[j * D128 + k];
    accR += e * w_rel[j * D128 + k];
  }
  accW += W_b[k];
  if (r < n_rel) rel_out[(size_t)r * D128 + k] = accR;

  red[k] = accW * a_w[k];
  __syncthreads();
  for (int s = 64; s > 0; s >>= 1) { if (k < s) red[k] += red[k + s]; __syncthreads(); }
  if (k == 0) s1[r] = red[0];
  __syncthreads();
  red[k] = accW * a_w[D128 + k];
  __syncthreads();
  for (int s = 64; s > 0; s >>= 1) { if (k < s) red[k] += red[k + s]; __syncthreads(); }
  if (k == 0) s2[r] = red[0];
}

// ---------------- per-edge alpha + degree -------
__global__ void k_edge(const int* __restrict__ row, const int* __restrict__ et,
                       const int* __restrict__ qt,
                       const float* __restrict__ s1, const float* __restrict__ s2,
                       const float* __restrict__ a_b,
                       float* __restrict__ alpha, float* __restrict__ deg,
                       float* __restrict__ sum_alpha, int E) {
  int e = blockIdx.x * blockDim.x + threadIdx.x;
  if (e >= E) return;
  float l = s1[et[e]] + s2[qt[e]] + a_b[0];
  float lr = l > 0.f ? l : NEG_SLOPE * l;
  float a = expf(lr);
  alpha[e] = a;
  int r = row[e];
  atomicAdd(&deg[r], 1.0f);
  atomicAdd(&sum_alpha[r], a);
}

// ---------------- edge scatter: agg[row] += (x_j * r_e) * coeff --------
// one wave32 per edge, 4 floats per lane (float4 gathers, L2-resident tables)
__global__ __launch_bounds__(256) void k_scatter(
    const int* __restrict__ row, const int* __restrict__ col,
    const int* __restrict__ et,
    const float* __restrict__ ent, const float* __restrict__ rel_embed,
    const float* __restrict__ alpha, const float* __restrict__ deg,
    const float* __restrict__ sum_alpha,
    float* __restrict__ agg, int E) {
  int gt = blockIdx.x * blockDim.x + threadIdx.x;
  int e = gt >> 5;
  int lane = gt & 31;
  if (e >= E) return;
  int r = row[e];
  float coeff = alpha[e] / (sum_alpha[r] * deg[r]);  // alpha-norm * 1/deg
  int c = col[e];
  int t = et[e];                                     // always < n_rel
  int d = lane << 2;
  const float4 xj = *(const float4*)(ent + (size_t)c * D128 + d);
  const float4 re = *(const float4*)(rel_embed + (size_t)t * D128 + d);
  float* dst = agg + (size_t)r * D128 + d;
  atomicAdd(dst + 0, xj.x * re.x * coeff);
  atomicAdd(dst + 1, xj.y * re.y * coeff);
  atomicAdd(dst + 2, xj.z * re.z * coeff);
  atomicAdd(dst + 3, xj.w * re.w * coeff);
}

// ---------------- fused WMMA GEMM (persistent blocks, B in registers) --
// out = agg @ wA + ent @ wB + bias      (0.5 and loop_rel pre-folded)
// block = 256 thr = 8 waves; wave w owns N-tile w (16 cols).
// Each wave holds its full B panels (K=128 x 16, both matrices) in 128 VGPRs
// and loops over M-tiles; inner loop is pure ds_load + v_wmma.
__global__ __launch_bounds__(256) void k_gemm(
    const float* __restrict__ agg, const float* __restrict__ ent,
    const float* __restrict__ wA, const float* __restrict__ wB,
    const float* __restrict__ bias,
    float* __restrict__ out, int N, int mtiles) {
  const int LDST = 132;                 // pad: stride 128 floats == 1 bank
  __shared__ float A0[16 * LDST];
  __shared__ float A1[16 * LDST];
  int tid = threadIdx.x;
  int wv = tid >> 5;
  int lane = tid & 31;
  int n0 = wv << 4;
  int m_in = lane & 15;
  int koff = (lane >> 4) << 1;          // A/B layout: VGPR0 = K=0|2, VGPR1 = K=1|3
  int ncol = lane & 15;                 // B/C/D: row striped across lanes
  int rbase = (lane >> 4) << 3;         // lanes 16-31 hold rows M=r+8

  // Preload both B panels into registers (reused across all M-tiles).
  v2f Bo[32], Bl[32];
#pragma unroll
  for (int i = 0; i < 32; ++i) {
    int kb = (i << 2) + koff;
    Bo[i].x = wA[kb * D128 + n0 + ncol];
    Bo[i].y = wA[(kb + 1) * D128 + n0 + ncol];
    Bl[i].x = wB[kb * D128 + n0 + ncol];
    Bl[i].y = wB[(kb + 1) * D128 + n0 + ncol];
  }
  float bv = bias[n0 + ncol];

  for (int mt = blockIdx.x; mt < mtiles; mt += gridDim.x) {
    int m0 = mt << 4;
    __syncthreads();                    // LDS reuse guard
    for (int i = tid; i < 16 * (D128 / 4); i += 256) {   // 512 float4s
      int rr = i >> 5;
      int k4 = (i & 31) << 2;
      int rowi = m0 + rr;
      float4 va = {0.f, 0.f, 0.f, 0.f}, vp = {0.f, 0.f, 0.f, 0.f};
      if (rowi < N) {
        va = *(const float4*)(agg + (size_t)rowi * D128 + k4);
        vp = *(const float4*)(ent + (size_t)rowi * D128 + k4);
      }
      *(float4*)(A0 + rr * LDST + k4) = va;
      *(float4*)(A1 + rr * LDST + k4) = vp;
    }
    __syncthreads();

    v8f acc = {};
#pragma unroll
    for (int i = 0; i < 32; ++i) {
      v2f a;
      a.x = A0[m_in * LDST + (i << 2) + koff];
      a.y = A0[m_in * LDST + (i << 2) + koff + 1];
      acc = __builtin_amdgcn_wmma_f32_16x16x4_f32(false, a, false, Bo[i],
                                                  (short)0, acc, false, false);
    }
#pragma unroll
    for (int i = 0; i < 32; ++i) {
      v2f a;
      a.x = A1[m_in * LDST + (i << 2) + koff];
      a.y = A1[m_in * LDST + (i << 2) + koff + 1];
      acc = __builtin_amdgcn_wmma_f32_16x16x4_f32(false, a, false, Bl[i],
                                                  (short)0, acc, false, false);
    }
#pragma unroll
    for (int rr = 0; rr < 8; ++rr) {
      int rowo = m0 + rr + rbase;
      if (rowo < N) out[(size_t)rowo * D128 + n0 + ncol] = acc[rr] + bv;
    }
  }
}

// ---------------- BatchNorm stats / finalize / apply ----------------
__global__ __launch_bounds__(128) void k_bnstats(
    const float* __restrict__ out, float* __restrict__ colsum,
    float* __restrict__ colsumsq, int N, int rpb) {
  int d = threadIdx.x;
  int r0 = blockIdx.x * rpb;
  float s = 0.f, ss = 0.f;
  for (int i = 0; i < rpb; ++i) {
    int r = r0 + i;
    if (r < N) { float v = out[(size_t)r * D128 + d]; s += v; ss += v * v; }
  }
  atomicAdd(&colsum[d], s);
  atomicAdd(&colsumsq[d], ss);
}

__global__ void k_bnfinal(const float* __restrict__ colsum,
                          const float* __restrict__ colsumsq,
                          const float* __restrict__ gamma,
                          const float* __restrict__ beta,
                          float* __restrict__ sc, float* __restrict__ sh, int N) {
  int d = threadIdx.x;
  float inv_n = 1.0f / (float)N;
  float mean = colsum[d] * inv_n;
  float var = colsumsq[d] * inv_n - mean * mean;   // biased var
  float g = rsqrtf(var + BN_EPS) * gamma[d];
  sc[d] = g;
  sh[d] = beta[d] - mean * g;
}

__global__ void k_bnapply(float* __restrict__ out, const float* __restrict__ sc,
                          const float* __restrict__ sh, int total) {
  int i = blockIdx.x * blockDim.x + threadIdx.x;
  if (i >= total) return;
  int d = i & 127;
  float y = out[i] * sc[d] + sh[d];
  out[i] = y > 0.f ? y : NEG_SLOPE * y;
}

extern "C" void kernel_launch(void* const* d_in, const int* in_sizes, int n_in,
                              void* d_out, int out_size, void* d_ws, size_t ws_size,
                              hipStream_t stream) {
  (void)n_in; (void)out_size; (void)ws_size;
  const int*   edge_index = (const int*)d_in[0];
  const int*   edge_type  = (const int*)d_in[1];
  const int*   query_type = (const int*)d_in[2];
  const float* ent        = (const float*)d_in[3];
  const float* rel        = (const float*)d_in[4];
  const float* w_loop     = (const float*)d_in[5];
  const float* w_out      = (const float*)d_in[6];
  const float* w_rel      = (const float*)d_in[7];
  const float* loop_rel   = (const float*)d_in[8];
  const float* W_w        = (const float*)d_in[9];
  const float* W_b        = (const float*)d_in[10];
  const float* a_w        = (const float*)d_in[11];
  const float* a_b        = (const float*)d_in[12];
  const float* bias       = (const float*)d_in[13];
  const float* gamma      = (const float*)d_in[14];
  const float* beta       = (const float*)d_in[15];

  const int E = in_sizes[1];
  const int N = in_sizes[3] / D128;
  const int R = in_sizes[4] / D128;

  const int* row = edge_index;
  const int* col = edge_index + E;

  float* ws = (float*)d_ws;
  size_t o = 0;
  float* agg       = ws + o; o += (size_t)N * D128;
  float* deg       = ws + o; o += N;
  float* sum_alpha = ws + o; o += N;
  float* alpha     = ws + o; o += E;
  float* s1        = ws + o; o += 512;
  float* s2        = ws + o; o += 512;
  float* colsum    = ws + o; o += 128;
  float* colsumsq  = ws + o; o += 128;
  float* sc        = ws + o; o += 128;
  float* sh        = ws + o; o += 128;
  int ztotal = (int)o;                       // region needing zeros
  float* wA        = ws + o; o += D128 * D128;
  float* wB        = ws + o; o += D128 * D128;

  float* ent_out = (float*)d_out;
  float* rel_out = ent_out + (size_t)N * D128;

  k_zero<<<(ztotal + 255) / 256, 256, 0, stream>>>(ws, ztotal);
  k_wprep<<<(D128 * D128 + 255) / 256, 256, 0, stream>>>(w_out, w_loop, loop_rel,
                                                         wA, wB);
  k_rel_prep<<<R + 1, 128, 0, stream>>>(rel, loop_rel, W_w, W_b, a_w, w_rel,
                                        s1, s2, rel_out, R);
  k_edge<<<(E + 255) / 256, 256, 0, stream>>>(row, edge_type, query_type,
                                              s1, s2, a_b, alpha, deg, sum_alpha, E);
  long long sthreads = (long long)E * 32;
  k_scatter<<<(unsigned)((sthreads + 255) / 256), 256, 0, stream>>>(
      row, col, edge_type, ent, rel, alpha, deg, sum_alpha, agg, E);
  int mtiles = (N + 15) / 16;
  int gblocks = mtiles < 640 ? mtiles : 640;  // persistent blocks
  k_gemm<<<gblocks, 256, 0, stream>>>(agg, ent, wA, wB, bias, ent_out, N, mtiles);
  const int RPB = 256;
  k_bnstats<<<(N + RPB - 1) / RPB, 128, 0, stream>>>(ent_out, colsum, colsumsq, N, RPB);
  k_bnfinal<<<1, 128, 0, stream>>>(colsum, colsumsq, gamma, beta, sc, sh, N);
  int total = N * D128;
  k_bnapply<<<(total + 255) / 256, 256, 0, stream>>>(ent_out, sc, sh, total);
}